// ProteinConv2D_31069793419621
// MI455X (gfx1250) — compile-verified
//
#include <hip/hip_runtime.h>

typedef __attribute__((ext_vector_type(2))) float v2f;
typedef __attribute__((ext_vector_type(8))) float v8f;

#define LL    128
#define LL2   256
#define PA    132                  // A pitch: gcd(132,64)=4 -> 16-lane conflict-free
#define PB    161                  // B pitch: gcd(161,64)=1, width need 159
#define TX    32                   // output x-tile per workgroup
#define BUFSZ (16 * PA + 16 * PB)  // 4688 floats per stage buffer
#define DUMMY (2 * BUFSZ)          // dummy LDS float for clamped async lanes

// out[b, i*8+j, y, x] = sum_{u,v} v1[b,i,u,v] * v2[b,j,u-(y-128), v-(x-128)]
__global__ __launch_bounds__(256) void pconv_xcorr_wmma(const float* __restrict__ v1,
                                                        const float* __restrict__ v2,
                                                        float* __restrict__ out) {
  __shared__ __align__(16) float lds[2 * BUFSZ + 1];

  const int x0 = blockIdx.x * TX;   // 0..224
  const int y  = blockIdx.y;        // 0..255
  const int p  = blockIdx.z;        // batch pair 0..3
  const int b0 = 2 * p;

  const int t    = threadIdx.x;
  const int wid  = t >> 5;          // wave 0..7
  const int lane = t & 31;

  const int dy   = y - LL;
  const int u_lo = dy > 0 ? dy : 0;
  const int u_hi = dy < 0 ? LL + dy : LL;

  // ---- zero both stage buffers once (B zero-padding is invariant per block) ----
  for (int e = t; e < 2 * BUFSZ + 1; e += 256) lds[e] = 0.0f;

  // ---- per-thread async-staging constants ----
  // A: thread t loads 8 floats (two b128) of row mA, cols cA..cA+7.
  const int mA = t >> 4;
  const int cA = (t & 15) * 8;
  const unsigned gArow = (unsigned)(((b0 + (mA >> 3)) * 8 + (mA & 7)) * (LL * LL));
  const unsigned lArel = (unsigned)((mA * PA + cA) * 4);

  // B: valid LDS idx window [i0,i1) holds v2 columns (idx + 97 - x0).
  const int i0v  = (x0 - 97) > 0 ? (x0 - 97) : 0;
  const int i1v  = (x0 + 31) < 160 ? (x0 + 31) : 160;
  const int lenB = i1v - i0v;       // <= 128
  unsigned gBrow[8], gBcol[8], lBrel[8];
  bool okB[8];
#pragma unroll
  for (int q = 0; q < 8; ++q) {
    const int e = t + 256 * q;
    const int r = e >> 7, cc = e & 127;
    const int idx = i0v + cc;
    okB[q]   = (cc < lenB);
    gBrow[q] = (unsigned)(((b0 + (r >> 3)) * 8 + (r & 7)) * (LL * LL));
    gBcol[q] = okB[q] ? (unsigned)(idx + 97 - x0) : 0u;
    lBrel[q] = (unsigned)((16 * PA + r * PB + idx) * 4);
  }

  const unsigned long long v1b = (unsigned long long)v1;
  const unsigned long long v2b = (unsigned long long)v2;

  // Issue exactly 10 async ops per thread per stage (uniform ASYNCcnt math).
  auto stage = [&](int u, int bufsel) {
    const unsigned bufbyte = (unsigned)(bufsel * BUFSZ * 4);
    const unsigned ga = (gArow + (unsigned)u * LL + (unsigned)cA) * 4u;
    const unsigned la = bufbyte + lArel;
    asm volatile("global_load_async_to_lds_b128 %0, %1, %2"
                 :: "v"(la), "v"(ga), "s"(v1b) : "memory");
    asm volatile("global_load_async_to_lds_b128 %0, %1, %2 offset:16"
                 :: "v"(la), "v"(ga), "s"(v1b) : "memory");
    const int ur = u - dy;          // v2 row, in [0,128)
#pragma unroll
    for (int q = 0; q < 8; ++q) {
      const unsigned gb = (gBrow[q] + (unsigned)ur * LL + gBcol[q]) * 4u;
      const unsigned lb = okB[q] ? (bufbyte + lBrel[q]) : (unsigned)(DUMMY * 4);
      asm volatile("global_load_async_to_lds_b32 %0, %1, %2"
                   :: "v"(lb), "v"(gb), "s"(v2b) : "memory");
    }
  };

  v8f acc0 = {}, acc1 = {}, acc2 = {}, acc3 = {};  // x = x0 + wid + {0,8,16,24}

  const int m16 = lane & 15;        // A row M / B row N
  const int k0  = (lane >> 4) << 1; // K pair base (0 or 2)
  const int sh0 = 31 - wid;         // LDS shift for x = x0 + wid

  __syncthreads();                  // buffers zeroed

  if (u_lo < u_hi) stage(u_lo, 0);

  int cur = 0;
  for (int u = u_lo; u < u_hi; ++u, cur ^= 1) {
    if (u + 1 < u_hi) {
      stage(u + 1, cur ^ 1);        // overlap next row's DMA with this compute
      asm volatile("s_wait_asynccnt 0xa" ::: "memory");   // cur's 10 ops done
    } else {
      asm volatile("s_wait_asynccnt 0x0" ::: "memory");
    }
    __syncthreads();

    const float* ldsA = lds + cur * BUFSZ;
    const float* ldsB = ldsA + 16 * PA;
#pragma unroll 4
    for (int v = 0; v < LL; v += 4) {
      v2f a = *(const v2f*)&ldsA[m16 * PA + v + k0];   // 8B-aligned
      const int bi = m16 * PB + v + k0 + sh0;
      v2f bv0, bv1, bv2, bv3;
      bv0.x = ldsB[bi];      bv0.y = ldsB[bi + 1];
      bv1.x = ldsB[bi - 8];  bv1.y = ldsB[bi - 7];
      bv2.x = ldsB[bi - 16]; bv2.y = ldsB[bi - 15];
      bv3.x = ldsB[bi - 24]; bv3.y = ldsB[bi - 23];
      acc0 = __builtin_amdgcn_wmma_f32_16x16x4_f32(false, a, false, bv0,
                                                   (short)0, acc0, false, false);
      acc1 = __builtin_amdgcn_wmma_f32_16x16x4_f32(false, a, false, bv1,
                                                   (short)0, acc1, false, false);
      acc2 = __builtin_amdgcn_wmma_f32_16x16x4_f32(false, a, false, bv2,
                                                   (short)0, acc2, false, false);
      acc3 = __builtin_amdgcn_wmma_f32_16x16x4_f32(false, a, false, bv3,
                                                   (short)0, acc3, false, false);
    }
    __syncthreads();
  }

  // ---- stage the 128x32 result tile in LDS (pitch 33), then coalesced stores ----
  // Useful D entries: lanes 0-7  -> batch b0:   i = r, j = lane
  //                   lanes 24-31-> batch b0+1: i = r, j = lane-24
  __syncthreads();
  if (lane < 8 || lane >= 24) {
    const int base = (lane < 8) ? lane : (64 + (lane - 24));
#pragma unroll
    for (int r = 0; r < 8; ++r) {
      float* o = &lds[(base + r * 8) * 33];
      o[wid]      = acc0[r];
      o[wid + 8]  = acc1[r];
      o[wid + 16] = acc2[r];
      o[wid + 24] = acc3[r];
    }
  }
  __syncthreads();
  {
    const int cp = t >> 1;            // channel-pair 0..127
    const int xt = (t & 1) * 16;      // 0 or 16
    const int b  = b0 + (cp >> 6);
    const int cc = cp & 63;
    float tmp[16];
#pragma unroll
    for (int q = 0; q < 16; ++q) tmp[q] = lds[cp * 33 + xt + q];
    float4* dst = (float4*)&out[(((b * 64 + cc) * LL2) + y) * LL2 + x0 + xt];
    dst[0] = make_float4(tmp[0],  tmp[1],  tmp[2],  tmp[3]);
    dst[1] = make_float4(tmp[4],  tmp[5],  tmp[6],  tmp[7]);
    dst[2] = make_float4(tmp[8],  tmp[9],  tmp[10], tmp[11]);
    dst[3] = make_float4(tmp[12], tmp[13], tmp[14], tmp[15]);
  }
}

extern "C" void kernel_launch(void* const* d_in, const int* in_sizes, int n_in,
                              void* d_out, int out_size, void* d_ws, size_t ws_size,
                              hipStream_t stream) {
  const float* v1 = (const float*)d_in[0];   // [8,8,128,128] f32
  const float* v2 = (const float*)d_in[1];   // [8,8,128,128] f32
  float* out = (float*)d_out;                // [8,64,256,256] f32

  dim3 grid(LL2 / TX, LL2, 4);               // (8, 256, 4)
  dim3 block(256);                           // 8 waves of 32
  pconv_xcorr_wmma<<<grid, block, 0, stream>>>(v1, v2, out);
}